// LSTMNet_36429912604796
// MI455X (gfx1250) — compile-verified
//
#include <hip/hip_runtime.h>
#include <hip/hip_bf16.h>
#include <cstdint>

// Problem shape (fixed by the reference): T=4096 steps, B=8192 batch, I=H=1.
#define T_STEPS 4096
#define BATCH   8192
#define BLOCK   256                  // 8 wave32s per workgroup
#define CHUNK   32                   // timesteps per staged LDS buffer (32KB each)
#define NCHUNK  (T_STEPS / CHUNK)    // 128
static_assert(T_STEPS % CHUNK == 0, "");

// ---- gfx1250 fast tanh (V_TANH_F32 is a CDNA5 transcendental) ----------------
__device__ __forceinline__ float fast_tanh(float x) {
#if __has_builtin(__builtin_amdgcn_tanhf)
  return __builtin_amdgcn_tanhf(x);
#else
  float r;
  asm("v_tanh_f32 %0, %1" : "=v"(r) : "v"(x));
  return r;
#endif
}

// ---- gfx1250 async global->LDS copies (ASYNCcnt-tracked) ---------------------
// VDST = per-lane LDS byte address, VADDR = per-lane 64-bit global address.
__device__ __forceinline__ void async_b128_to_lds(uint32_t lds_byte_off,
                                                  const float* gaddr) {
  asm volatile("global_load_async_to_lds_b128 %0, %1, off"
               :: "v"(lds_byte_off), "v"((uint64_t)(uintptr_t)gaddr)
               : "memory");
}

template <int N>
__device__ __forceinline__ void wait_asynccnt() {
#if __has_builtin(__builtin_amdgcn_s_wait_asynccnt)
  __builtin_amdgcn_s_wait_asynccnt(N);
#else
  asm volatile("s_wait_asynccnt %0" :: "n"(N) : "memory");
#endif
  asm volatile("" ::: "memory");  // keep LDS reads below the wait
}

// Stage one CHUNK of x into LDS buffer `buf` with 8 async B128 ops.
// Lane mapping keeps every wave's writes inside its own 32-column stripe,
// which is exactly the stripe that wave later reads -> per-wave ASYNCcnt
// ordering is sufficient, no block barrier needed.
__device__ __forceinline__ void stage_chunk(const float* gsrc0_lane,  // lane base for chunk 0, j=0
                                            uint32_t lds0_lane,      // lane LDS byte base (buf 0, j=0)
                                            int k, int buf) {
  const float*  g = gsrc0_lane + (size_t)k * (CHUNK * BATCH);
  const uint32_t l = lds0_lane + (uint32_t)buf * (CHUNK * BLOCK * 4);
#pragma unroll
  for (int j = 0; j < 8; ++j)
    async_b128_to_lds(l + (uint32_t)j * (4 * BLOCK * 4),   // +4 ts rows in LDS
                      g + (size_t)j * (4 * BATCH));        // +4 ts rows in memory
}

__global__ __launch_bounds__(BLOCK) void lstm_h1_kernel(
    const float* __restrict__ x,     // [T,B,1]
    const float* __restrict__ h0,    // [1,B,1]
    const float* __restrict__ c0,    // [1,B,1]
    const float* __restrict__ W_ih,  // [4,1]  gate order i,f,g,o
    const float* __restrict__ W_hh,  // [4,1]
    const float* __restrict__ b_ih,  // [4]
    const float* __restrict__ b_hh,  // [4]
    float* __restrict__ out)         // [T*B] out, then [B] hn, then [B] cn
{
  __shared__ float xbuf[3][CHUNK * BLOCK];   // 3 x 32KB = 96KB of the 320KB WGP LDS

  const int tid = threadIdx.x;
  const int b0  = blockIdx.x * BLOCK;
  const int b   = b0 + tid;                  // this lane's batch element

  // ---- per-lane staging geometry (wave-local B128 tiles) --------------------
  const int wv  = tid >> 5;                  // wave index within block (0..7)
  const int ln  = tid & 31;                  // lane within wave
  const int r   = ln >> 3;                   // ts row within a group of 4 (0..3)
  const int col = (wv << 5) + ((ln & 7) << 2);  // column quad start, 16B aligned

  const uint32_t lds_base = (uint32_t)(uintptr_t)&xbuf[0][0];
  const float*   gsrc0    = x + (size_t)r * BATCH + b0 + col;          // chunk0,j0
  const uint32_t lds0     = lds_base + (uint32_t)(r * BLOCK + col) * 4u;

  // ---- uniform gate parameters in SGPRs -------------------------------------
  // sigmoid(x) = 0.5*tanh(x/2)+0.5 : pre-halve i/f/o gate params so the only
  // h-dependent op before v_tanh is a single FMA (shortest serial chain).
  const float wi_i = 0.5f * W_ih[0], wi_f = 0.5f * W_ih[1];
  const float wi_g =        W_ih[2], wi_o = 0.5f * W_ih[3];
  const float wh_i = 0.5f * W_hh[0], wh_f = 0.5f * W_hh[1];
  const float wh_g =        W_hh[2], wh_o = 0.5f * W_hh[3];
  const float b_i  = 0.5f * (b_ih[0] + b_hh[0]);
  const float b_f  = 0.5f * (b_ih[1] + b_hh[1]);
  const float b_g  =        (b_ih[2] + b_hh[2]);
  const float b_o  = 0.5f * (b_ih[3] + b_hh[3]);

  float h = h0[b];
  float c = c0[b];

  // ---- prime two chunks (prefetch distance ~2*CHUNK steps) ------------------
  stage_chunk(gsrc0, lds0, 0, 0);
  stage_chunk(gsrc0, lds0, 1, 1);

#pragma unroll 1
  for (int k = 0; k < NCHUNK; ++k) {
    const int buf = k % 3;

    if (k + 2 < NCHUNK) {
      stage_chunk(gsrc0, lds0, k + 2, (k + 2) % 3);
      wait_asynccnt<16>();          // <=2 chunks outstanding -> chunk k done
    } else if (k + 1 < NCHUNK) {
      wait_asynccnt<8>();           // only chunk k+1 may be outstanding
    } else {
      wait_asynccnt<0>();
    }

    // Serial recurrence. Per-step dependency chain (7 ops):
    //   h -> v_fma -> v_tanh -> v_fma -> v_fma(c) -> ... -> v_tanh(c) -> v_mul
    // All xt-dependent work is off-chain and computable ahead of time.
#pragma unroll
    for (int ts = 0; ts < CHUNK; ++ts) {
      const float xt = xbuf[buf][ts * BLOCK + tid];

      const float ai = fmaf(xt, wi_i, b_i);   // off-chain (xt known early)
      const float af = fmaf(xt, wi_f, b_f);
      const float ag = fmaf(xt, wi_g, b_g);
      const float ao = fmaf(xt, wi_o, b_o);

      const float ti = fast_tanh(fmaf(h, wh_i, ai));
      const float tf = fast_tanh(fmaf(h, wh_f, af));
      const float tg = fast_tanh(fmaf(h, wh_g, ag));
      const float to = fast_tanh(fmaf(h, wh_o, ao));

      const float gi = fmaf(0.5f, ti, 0.5f);  // sigmoid(i)
      const float gf = fmaf(0.5f, tf, 0.5f);  // sigmoid(f)
      const float go = fmaf(0.5f, to, 0.5f);  // sigmoid(o)

      c = fmaf(gf, c, gi * tg);
      h = go * fast_tanh(c);

      // Stream-out: 128MB never re-read; keep it out of the caches.
      __builtin_nontemporal_store(h, out + (size_t)(k * CHUNK + ts) * BATCH + b);
    }
  }

  // hn, cn (flattened after out in return order)
  out[(size_t)T_STEPS * BATCH + b]         = h;
  out[(size_t)T_STEPS * BATCH + BATCH + b] = c;
}

extern "C" void kernel_launch(void* const* d_in, const int* in_sizes, int n_in,
                              void* d_out, int out_size, void* d_ws, size_t ws_size,
                              hipStream_t stream) {
  (void)in_sizes; (void)n_in; (void)out_size; (void)d_ws; (void)ws_size;

  const float* x    = (const float*)d_in[0];
  const float* h0   = (const float*)d_in[1];
  const float* c0   = (const float*)d_in[2];
  const float* W_ih = (const float*)d_in[3];
  const float* W_hh = (const float*)d_in[4];
  const float* b_ih = (const float*)d_in[5];
  const float* b_hh = (const float*)d_in[6];
  float* out = (float*)d_out;

  lstm_h1_kernel<<<dim3(BATCH / BLOCK), dim3(BLOCK), 0, stream>>>(
      x, h0, c0, W_ih, W_hh, b_ih, b_hh, out);
}